// SDARSimpleMoE_2886218023002
// MI455X (gfx1250) — compile-verified
//
#include <hip/hip_runtime.h>

// Problem sizes (fixed by the reference): B=4, S=1024 -> T=4096 tokens.
#define T_TOK 4096
#define DDIM  1024
#define NEXP  16
#define FDIM  512
#define MT    32      // tokens per expert tile (two WMMA M-tiles)

typedef __attribute__((ext_vector_type(2))) float v2f;
typedef __attribute__((ext_vector_type(8))) float v8f;

// CDNA5 fp32 tensor op: D(16x16,f32) = A(16x4,f32) * B(4x16,f32) + C
__device__ __forceinline__ v8f wmma_f32(v2f a, v2f b, v8f c) {
  return __builtin_amdgcn_wmma_f32_16x16x4_f32(
      /*neg_a=*/false, a, /*neg_b=*/false, b,
      /*c_mod=*/(short)0, c, /*reuse_a=*/false, /*reuse_b=*/false);
}

__device__ __forceinline__ v2f ld2(const float* __restrict__ p) {
  v2f r; r.x = p[0]; r.y = p[1]; return r;
}

// ---------------------------------------------------------------------------
// Kernel 0: clear the (poisoned) output buffer; it is accumulated atomically.
// ---------------------------------------------------------------------------
__global__ void zero_out_kernel(float4* __restrict__ out, int n4) {
  int i = blockIdx.x * blockDim.x + threadIdx.x;
  if (i < n4) out[i] = make_float4(0.f, 0.f, 0.f, 0.f);
}

// ---------------------------------------------------------------------------
// Kernel 1: router. One block per token. 16 threads per expert dot-product,
// fp32 softmax + top-2 + renormalize (softmax denominator cancels).
// ---------------------------------------------------------------------------
__global__ __launch_bounds__(256) void router_kernel(
    const float* __restrict__ x, const float* __restrict__ rw,
    int* __restrict__ topk_idx, float* __restrict__ topk_w) {
  __shared__ float red[NEXP][17];   // padded to dodge bank conflicts
  __shared__ float logits[NEXP];
  const int t   = blockIdx.x;
  const int tid = threadIdx.x;
  const int e   = tid >> 4;
  const int j   = tid & 15;

  const float* xr = x  + (size_t)t * DDIM;
  const float* wr = rw + (size_t)e * DDIM;
  float p = 0.0f;
  for (int d = j; d < DDIM; d += 16) p += xr[d] * wr[d];
  red[e][j] = p;
  __syncthreads();
  if (j == 0) {
    float s = 0.0f;
    #pragma unroll
    for (int q = 0; q < 16; ++q) s += red[e][q];
    logits[e] = s;
  }
  __syncthreads();
  if (tid == 0) {
    float mx = logits[0];
    #pragma unroll
    for (int q = 1; q < NEXP; ++q) mx = fmaxf(mx, logits[q]);
    float pr[NEXP];
    #pragma unroll
    for (int q = 0; q < NEXP; ++q) pr[q] = __expf(logits[q] - mx);
    // top-2, lowest index wins ties (matches jax.lax.top_k)
    int i0 = 0;
    #pragma unroll
    for (int q = 1; q < NEXP; ++q) if (pr[q] > pr[i0]) i0 = q;
    int i1 = (i0 == 0) ? 1 : 0;
    #pragma unroll
    for (int q = 0; q < NEXP; ++q) if (q != i0 && pr[q] > pr[i1]) i1 = q;
    const float ws = pr[i0] + pr[i1];   // softmax denom cancels in renorm
    topk_idx[2 * t + 0] = i0;
    topk_idx[2 * t + 1] = i1;
    topk_w[2 * t + 0] = pr[i0] / ws;
    topk_w[2 * t + 1] = pr[i1] / ws;
  }
}

// ---------------------------------------------------------------------------
// Kernel 2: deterministic expert-grouped token lists (block-wide prefix scan,
// no atomics -> stable ordering). One block per expert.
// ---------------------------------------------------------------------------
__global__ __launch_bounds__(256) void build_lists_kernel(
    const int* __restrict__ topk_idx, const float* __restrict__ topk_w,
    int* __restrict__ list, float* __restrict__ wlist, int* __restrict__ count) {
  __shared__ int scan[256];
  const int e   = blockIdx.x;
  const int tid = threadIdx.x;
  const int per = T_TOK / 256;        // 16
  const int t0  = tid * per;

  int local = 0;
  for (int i = 0; i < per; ++i) {
    int t = t0 + i;
    if (topk_idx[2 * t] == e || topk_idx[2 * t + 1] == e) local++;
  }
  scan[tid] = local;
  __syncthreads();
  for (int off = 1; off < 256; off <<= 1) {    // Hillis-Steele inclusive scan
    int v = (tid >= off) ? scan[tid - off] : 0;
    __syncthreads();
    scan[tid] += v;
    __syncthreads();
  }
  int pos = scan[tid] - local;                  // exclusive prefix
  for (int i = 0; i < per; ++i) {
    int t = t0 + i;
    int s = -1;
    if (topk_idx[2 * t] == e) s = 0;
    else if (topk_idx[2 * t + 1] == e) s = 1;
    if (s >= 0) {
      list[e * T_TOK + pos]  = t;
      wlist[e * T_TOK + pos] = topk_w[2 * t + s];
      pos++;
    }
  }
  if (tid == 255) count[e] = scan[255];
}

// ---------------------------------------------------------------------------
// Kernel 3: expert SwiGLU + down-proj on WMMA, distance-2 software pipeline.
// Grid: (T/MT token tiles, E experts); 256 threads = 8 wave32.
// MT=32 -> two 16-row M-tiles share every B load (2x arithmetic intensity).
// Two operand slices are kept in flight: WMMAs for slice k overlap the loads
// of slices k+4 and k+8, so each wait is a partial s_wait_loadcnt against
// loads issued two batches earlier (~2x L2 latency tolerance vs distance 1).
//
// fp32 WMMA per-lane layout (ISA 7.12.2, wave32):
//   A 16x4 : lane (h=lane/16, lr=lane%16): a[v] = A[lr][2h+v]
//   B 4x16 : b[v] = B[2h+v][lr]
//   C 16x16: c[r] -> row (r + 8h), col lr
// ---------------------------------------------------------------------------
__global__ __launch_bounds__(256) void moe_expert_kernel(
    const float* __restrict__ x,
    const float* __restrict__ gate_w,
    const float* __restrict__ up_w,
    const float* __restrict__ down_w,
    const int* __restrict__ list,
    const float* __restrict__ wlist,
    const int* __restrict__ count,
    float* __restrict__ out) {
  constexpr int HS_STRIDE = FDIM + 4;           // 516 % 64 banks == 4 -> conflict-free
  __shared__ float hs[MT * HS_STRIDE];          // 32 x 516 f32 = ~66 KB (<< 320 KB/WGP)
  __shared__ int   tok[MT];
  __shared__ float twt[MT];

  const int e   = blockIdx.y;
  const int cnt = count[e];
  const int m0  = blockIdx.x * MT;
  if (m0 >= cnt) return;
  const int nvalid = min(MT, cnt - m0);

  const int tid = threadIdx.x;
  if (tid < MT) {
    const bool valid = tid < nvalid;
    tok[tid] = valid ? list[e * T_TOK + m0 + tid] : list[e * T_TOK + m0];
    twt[tid] = valid ? wlist[e * T_TOK + m0 + tid] : 0.0f;
  }
  __syncthreads();

  const int wave = tid >> 5;        // 0..7
  const int lane = tid & 31;
  const int half = lane >> 4;       // 0/1
  const int lr   = lane & 15;

  const float* ge = gate_w + (size_t)e * FDIM * DDIM;
  const float* ue = up_w   + (size_t)e * FDIM * DDIM;
  const float* de = down_w + (size_t)e * DDIM * FDIM;

  // A-matrix row pointers for the two M-tiles (rows lr and 16+lr)
  const float* a0p = x + (size_t)tok[lr]      * DDIM + 2 * half;
  const float* a1p = x + (size_t)tok[16 + lr] * DDIM + 2 * half;

  // ---------------- GEMM1: gate & up, wave owns F-columns [wave*64, wave*64+64)
  const int n0 = wave * 64;
  const float* gp[4];
  const float* uq[4];
  #pragma unroll
  for (int q = 0; q < 4; ++q) {
    const int f = n0 + q * 16 + lr;
    gp[q] = ge + (size_t)f * DDIM + 2 * half;
    uq[q] = ue + (size_t)f * DDIM + 2 * half;
  }

  v8f ag0[4], ag1[4], au0[4], au1[4];
  {
    v8f z = {};
    #pragma unroll
    for (int q = 0; q < 4; ++q) { ag0[q] = z; ag1[q] = z; au0[q] = z; au1[q] = z; }
  }

  struct Slice1 { v2f a0, a1, bg[4], bu[4]; };
  auto load1 = [&](int k) {
    Slice1 s;
    s.a0 = ld2(a0p + k);
    s.a1 = ld2(a1p + k);
    #pragma unroll
    for (int q = 0; q < 4; ++q) { s.bg[q] = ld2(gp[q] + k); s.bu[q] = ld2(uq[q] + k); }
    return s;
  };
  auto mma1 = [&](const Slice1& s) {
    #pragma unroll
    for (int q = 0; q < 4; ++q) {
      ag0[q] = wmma_f32(s.a0, s.bg[q], ag0[q]);
      ag1[q] = wmma_f32(s.a1, s.bg[q], ag1[q]);
      au0[q] = wmma_f32(s.a0, s.bu[q], au0[q]);
      au1[q] = wmma_f32(s.a1, s.bu[q], au1[q]);
    }
  };

  // distance-2 pipeline over 256 k-slices
  Slice1 s0 = load1(0);
  Slice1 s1 = load1(4);
  for (int k = 8; k < DDIM; k += 8) {
    Slice1 t0 = load1(k);       // prefetch slice k
    mma1(s0);                   // compute slice k-8
    s0 = t0;
    Slice1 t1 = load1(k + 4);   // prefetch slice k+4
    mma1(s1);                   // compute slice k-4
    s1 = t1;
  }
  mma1(s0);
  mma1(s1);

  // fused SwiGLU in registers -> LDS (both M-tiles)
  #pragma unroll
  for (int q = 0; q < 4; ++q) {
    #pragma unroll
    for (int r = 0; r < 8; ++r) {
      const int col = n0 + q * 16 + lr;
      {
        const float g = ag0[q][r], u = au0[q][r];
        hs[(r + 8 * half) * HS_STRIDE + col] = (g / (1.0f + __expf(-g))) * u;
      }
      {
        const float g = ag1[q][r], u = au1[q][r];
        hs[(16 + r + 8 * half) * HS_STRIDE + col] = (g / (1.0f + __expf(-g))) * u;
      }
    }
  }
  __syncthreads();

  // ---------------- GEMM2: down-proj, 64 D-tiles; wave owns 8, paired 2 at a
  // time so both M-tile A operands feed 4 WMMAs per 4 loads.
  const float* h0 = hs + lr * HS_STRIDE + 2 * half;
  const float* h1 = hs + (16 + lr) * HS_STRIDE + 2 * half;

  for (int qt = 0; qt < 4; ++qt) {
    const int ndA = (wave * 8 + 2 * qt) * 16;
    const int ndB = ndA + 16;
    const float* dpA = de + (size_t)(ndA + lr) * FDIM + 2 * half;
    const float* dpB = de + (size_t)(ndB + lr) * FDIM + 2 * half;

    v8f accA0 = {}, accA1 = {}, accB0 = {}, accB1 = {};

    struct Slice2 { v2f bA, bB, a0, a1; };
    auto load2 = [&](int k) {
      Slice2 s;
      s.bA = ld2(dpA + k);
      s.bB = ld2(dpB + k);
      s.a0 = ld2(h0 + k);
      s.a1 = ld2(h1 + k);
      return s;
    };
    auto mma2 = [&](const Slice2& s) {
      accA0 = wmma_f32(s.a0, s.bA, accA0);
      accA1 = wmma_f32(s.a1, s.bA, accA1);
      accB0 = wmma_f32(s.a0, s.bB, accB0);
      accB1 = wmma_f32(s.a1, s.bB, accB1);
    };

    Slice2 s0d = load2(0);
    Slice2 s1d = load2(4);
    for (int k = 8; k < FDIM; k += 8) {
      Slice2 t0 = load2(k);
      mma2(s0d);
      s0d = t0;
      Slice2 t1 = load2(k + 4);
      mma2(s1d);
      s1d = t1;
    }
    mma2(s0d);
    mma2(s1d);

    #pragma unroll
    for (int r = 0; r < 8; ++r) {
      const int row0 = r + 8 * half;
      const int row1 = 16 + r + 8 * half;
      if (row0 < nvalid) {
        const float w0 = twt[row0];
        float* o0 = out + (size_t)tok[row0] * DDIM + lr;
        atomicAdd(o0 + ndA, accA0[r] * w0);
        atomicAdd(o0 + ndB, accB0[r] * w0);
      }
      if (row1 < nvalid) {
        const float w1 = twt[row1];
        float* o1 = out + (size_t)tok[row1] * DDIM + lr;
        atomicAdd(o1 + ndA, accA1[r] * w1);
        atomicAdd(o1 + ndB, accB1[r] * w1);
      }
    }
  }
}

// ---------------------------------------------------------------------------
extern "C" void kernel_launch(void* const* d_in, const int* in_sizes, int n_in,
                              void* d_out, int out_size, void* d_ws, size_t ws_size,
                              hipStream_t stream) {
  (void)in_sizes; (void)n_in; (void)out_size; (void)ws_size;
  const float* x        = (const float*)d_in[0];   // [4,1024,1024] fp32
  const float* router_w = (const float*)d_in[1];   // [16,1024]
  const float* gate_w   = (const float*)d_in[2];   // [16,512,1024]
  const float* up_w     = (const float*)d_in[3];   // [16,512,1024]
  const float* down_w   = (const float*)d_in[4];   // [16,1024,512]
  float* out = (float*)d_out;                      // [4,1024,1024] fp32

  // Workspace layout (~576 KB)
  char* w = (char*)d_ws;
  int*   topk_idx = (int*)w;    w += (size_t)T_TOK * 2 * sizeof(int);
  float* topk_w   = (float*)w;  w += (size_t)T_TOK * 2 * sizeof(float);
  int*   list     = (int*)w;    w += (size_t)NEXP * T_TOK * sizeof(int);
  float* wlist    = (float*)w;  w += (size_t)NEXP * T_TOK * sizeof(float);
  int*   count    = (int*)w;    w += (size_t)NEXP * sizeof(int);

  const int n4 = (T_TOK * DDIM) / 4;
  zero_out_kernel<<<(n4 + 255) / 256, 256, 0, stream>>>((float4*)out, n4);
  router_kernel<<<T_TOK, 256, 0, stream>>>(x, router_w, topk_idx, topk_w);
  build_lists_kernel<<<NEXP, 256, 0, stream>>>(topk_idx, topk_w, list, wlist, count);

  dim3 grid(T_TOK / MT, NEXP);
  moe_expert_kernel<<<grid, 256, 0, stream>>>(x, gate_w, up_w, down_w,
                                              list, wlist, count, out);
}